// MLPBlockCore_55405078118646
// MI455X (gfx1250) — compile-verified
//
#include <hip/hip_runtime.h>
#include <math.h>

typedef __attribute__((ext_vector_type(2))) float v2f;
typedef __attribute__((ext_vector_type(8))) float v8f;

#define ALPHA 1.702f
#define LIMIT 7.0f
// T=512, H=512, E=16, K=2, I=256, 2I=512

// ---------------- Router: logits -> top2 -> softmax ----------------
// One wave per token. Lane l: expert e=l&15, H-half h0=(l>>4)*256.
__global__ __launch_bounds__(128)
void router_kernel(const float* __restrict__ normed,
                   const float* __restrict__ gk,   // [512,16]
                   const float* __restrict__ gb,   // [16]
                   int* __restrict__ idx,          // [512*2]
                   float* __restrict__ wgt) {      // [512*2]
  const int wave = threadIdx.x >> 5;
  const int lane = threadIdx.x & 31;
  const int t = blockIdx.x * 4 + wave;            // grid=128 -> t<512 exact
  const int e = lane & 15;
  const int h0 = (lane >> 4) * 256;
  const float* x = normed + (size_t)t * 512;
  float acc = 0.f;
#pragma unroll 8
  for (int h = 0; h < 256; ++h)
    acc = fmaf(x[h0 + h], gk[(h0 + h) * 16 + e], acc);
  acc += __shfl_xor(acc, 16);                     // combine the two H halves
  float logit = acc + gb[e];                      // valid on lanes 0..15 (dup on 16..31)
  // top-2 (all lanes compute identically; shfl must stay uniform)
  float best = -3.4e38f, second = -3.4e38f;
  int bi = 0, si = 0;
  for (int j = 0; j < 16; ++j) {
    float v = __shfl(logit, j);
    if (v > best)        { second = best; si = bi; best = v; bi = j; }
    else if (v > second) { second = v; si = j; }
  }
  if (lane == 0) {
    float w0 = __builtin_amdgcn_rcpf(1.0f + __expf(second - best)); // softmax {best,second}
    idx[t * 2] = bi;  idx[t * 2 + 1] = si;
    wgt[t * 2] = w0;  wgt[t * 2 + 1] = 1.0f - w0;
  }
}

// ---------------- Bucket build: group (token,k) pairs by expert ----------------
__global__ void bucket_kernel(const int* __restrict__ idx,
                              int* __restrict__ bucket,   // [1024] pair ids grouped by expert
                              int* __restrict__ offs) {   // [17]
  if (threadIdx.x == 0 && blockIdx.x == 0) {
    int cnt[16];
    for (int e = 0; e < 16; ++e) cnt[e] = 0;
    for (int p = 0; p < 1024; ++p) cnt[idx[p]]++;
    int off = 0;
    for (int e = 0; e < 16; ++e) { offs[e] = off; off += cnt[e]; cnt[e] = 0; }
    offs[16] = off;                               // == 1024
    for (int p = 0; p < 1024; ++p) {
      int e = idx[p];
      bucket[offs[e] + cnt[e]++] = p;             // p = token*2 + k, token order preserved
    }
  }
}

// ---------------- MLP1 grouped GEMM + fused swiglu ----------------
// grid = (expert e, nTile of 2I=512/64); block = 4 waves, wave w owns 16 columns.
// Per 16-row M tile: X rows (gathered tokens) staged in LDS; K-loop over H=512 via
// v_wmma_f32_16x16x4_f32. Swiglu pairs interleaved columns with shfl_xor(1).
__global__ __launch_bounds__(128)
void moe_mlp1_kernel(const float* __restrict__ normed,
                     const float* __restrict__ w1,   // [16,512,512]
                     const float* __restrict__ b1,   // [16,512]
                     const int* __restrict__ bucket,
                     const int* __restrict__ offs,
                     float* __restrict__ act) {      // [1024,256]
  __shared__ float Xs[16 * 512];                     // 32 KB
  __shared__ int pids[16];
  const int e = blockIdx.x;
  const int tid = threadIdx.x;
  const int lane = tid & 31;
  const int nBase = blockIdx.y * 64 + (tid >> 5) * 16;
  const int col = nBase + (lane & 15);               // output column in [0,512)
  const int kk = (lane >> 4) * 2;                    // K pair owned by this lane half
  const int start = offs[e], end = offs[e + 1];
  const float* w1e = w1 + (size_t)e * 512 * 512;
  const float* bp = w1e + (size_t)col * 512 + kk;    // K-dim contiguous (row-major [out][h])
  const float bias = b1[e * 512 + col];
  const int gr = tid >> 3;                           // gather: row 0..15
  const int gc = (tid & 7) * 64;                     // 64-float chunk per thread
  const bool evenCol = ((col & 1) == 0);

  for (int m0 = start; m0 < end; m0 += 16) {
    __syncthreads();                                 // protect prev-iter Xs/pids readers
    if (tid < 16) {
      int r = m0 + tid;
      pids[tid] = bucket[(r < end) ? r : m0];        // pad with a valid row
    }
    __syncthreads();
    {
      const int tok = pids[gr] >> 1;
      const float4* src = (const float4*)(normed + (size_t)tok * 512 + gc);
      float4* dst = (float4*)(Xs + gr * 512 + gc);
#pragma unroll
      for (int i = 0; i < 16; ++i) dst[i] = src[i];
    }
    __syncthreads();

    v8f c = {};
    const float* ap = Xs + (lane & 15) * 512 + kk;   // A: lanes0-15 K={0,1}, 16-31 K={2,3}
#pragma unroll 8
    for (int h = 0; h < 512; h += 4) {
      v2f a = *(const v2f*)(ap + h);                 // ds_load_b64
      v2f b = *(const v2f*)(bp + h);                 // global_load_b64
      c = __builtin_amdgcn_wmma_f32_16x16x4_f32(false, a, false, b, (short)0, c,
                                                false, false);
    }
    // C layout: VGPR r -> M = r + 8*(lane>>4), N = lane&15
#pragma unroll
    for (int r = 0; r < 8; ++r) {
      float v = c[r] + bias;
      float p = __shfl_xor(v, 1);                    // partner interleaved column
      float gate = evenCol ? v : p;                  // x[..., ::2]
      float lin  = evenCol ? p : v;                  // x[..., 1::2]
      gate = fminf(gate, LIMIT);
      lin  = fminf(fmaxf(lin, -LIMIT), LIMIT);
      // gate * sigmoid(ALPHA*gate) * (lin+1), native v_rcp for the sigmoid
      float sig = __builtin_amdgcn_rcpf(1.0f + __expf(-ALPHA * gate));
      float a = gate * sig * (lin + 1.0f);
      int mrow = r + ((lane >> 4) << 3);
      if (evenCol && (m0 + mrow) < end)
        act[(size_t)pids[mrow] * 256 + (col >> 1)] = a;
    }
  }
}

// ---------------- MLP2 grouped GEMM + router weight ----------------
__global__ __launch_bounds__(128)
void moe_mlp2_kernel(const float* __restrict__ act,  // [1024,256]
                     const float* __restrict__ w2,   // [16,512,256]
                     const float* __restrict__ b2,   // [16,512]
                     const int* __restrict__ bucket,
                     const int* __restrict__ offs,
                     const float* __restrict__ wgt,
                     float* __restrict__ eo) {       // [1024,512]
  __shared__ float As[16 * 256];                     // 16 KB
  __shared__ int pids[16];
  const int e = blockIdx.x;
  const int tid = threadIdx.x;
  const int lane = tid & 31;
  const int nBase = blockIdx.y * 64 + (tid >> 5) * 16;
  const int col = nBase + (lane & 15);               // output column in [0,512)
  const int kk = (lane >> 4) * 2;
  const int start = offs[e], end = offs[e + 1];
  const float* w2e = w2 + (size_t)e * 512 * 256;
  const float* bp = w2e + (size_t)col * 256 + kk;    // K-dim (I) contiguous
  const float bias = b2[e * 512 + col];
  const int gr = tid >> 3;
  const int gc = (tid & 7) * 32;

  for (int m0 = start; m0 < end; m0 += 16) {
    __syncthreads();
    if (tid < 16) {
      int r = m0 + tid;
      pids[tid] = bucket[(r < end) ? r : m0];
    }
    __syncthreads();
    {
      const int pid = pids[gr];
      const float4* src = (const float4*)(act + (size_t)pid * 256 + gc);
      float4* dst = (float4*)(As + gr * 256 + gc);
#pragma unroll
      for (int i = 0; i < 8; ++i) dst[i] = src[i];
    }
    __syncthreads();

    v8f c = {};
    const float* ap = As + (lane & 15) * 256 + kk;
#pragma unroll 8
    for (int h = 0; h < 256; h += 4) {
      v2f a = *(const v2f*)(ap + h);
      v2f b = *(const v2f*)(bp + h);
      c = __builtin_amdgcn_wmma_f32_16x16x4_f32(false, a, false, b, (short)0, c,
                                                false, false);
    }
#pragma unroll
    for (int r = 0; r < 8; ++r) {
      int mrow = r + ((lane >> 4) << 3);
      if ((m0 + mrow) < end) {
        int pid = pids[mrow];
        eo[(size_t)pid * 512 + col] = (c[r] + bias) * wgt[pid];
      }
    }
  }
}

// ---------------- Combine the two expert contributions per token ----------------
__global__ __launch_bounds__(256)
void combine_kernel(const float* __restrict__ eo, float* __restrict__ out) {
  int gi = blockIdx.x * 256 + threadIdx.x;  // float4 index; 512*512/4 = 65536 total
  int t = gi >> 7;                          // 128 float4 per token row
  int h4 = gi & 127;
  const float4* e0 = (const float4*)eo + (size_t)(2 * t) * 128 + h4;
  const float4* e1 = (const float4*)eo + (size_t)(2 * t + 1) * 128 + h4;
  float4 a = *e0, b = *e1;
  float4 r;
  r.x = a.x + b.x; r.y = a.y + b.y; r.z = a.z + b.z; r.w = a.w + b.w;
  ((float4*)out)[gi] = r;
}

extern "C" void kernel_launch(void* const* d_in, const int* in_sizes, int n_in,
                              void* d_out, int out_size, void* d_ws, size_t ws_size,
                              hipStream_t stream) {
  const float* normed = (const float*)d_in[0];
  const float* gk     = (const float*)d_in[1];
  const float* gb     = (const float*)d_in[2];
  const float* w1     = (const float*)d_in[3];
  const float* b1     = (const float*)d_in[4];
  const float* w2     = (const float*)d_in[5];
  const float* b2     = (const float*)d_in[6];
  float* out = (float*)d_out;

  char* ws = (char*)d_ws;
  int*   idx    = (int*)(ws + 0);                       // 1024 ints
  float* wgt    = (float*)(ws + 4096);                  // 1024 floats
  int*   bucket = (int*)(ws + 8192);                    // 1024 ints
  int*   offs   = (int*)(ws + 12288);                   // 17 ints
  float* act    = (float*)(ws + 16384);                 // 1024*256 floats (1 MB)
  float* eo     = (float*)(ws + 16384 + 1024 * 256 * 4);// 1024*512 floats (2 MB)

  router_kernel<<<128, 128, 0, stream>>>(normed, gk, gb, idx, wgt);
  bucket_kernel<<<1, 32, 0, stream>>>(idx, bucket, offs);
  moe_mlp1_kernel<<<dim3(16, 8), 128, 0, stream>>>(normed, w1, b1, bucket, offs, act);
  moe_mlp2_kernel<<<dim3(16, 8), 128, 0, stream>>>(act, w2, b2, bucket, offs, wgt, eo);
  combine_kernel<<<256, 256, 0, stream>>>(eo, out);
}